// InvariantCrossAttention_49100066128121
// MI455X (gfx1250) — compile-verified
//
#include <hip/hip_runtime.h>
#include <math.h>

// ---------------------------------------------------------------------------
// Invariant cross attention, fused per-l workgroup kernel for gfx1250.
// Algebraic folding removes the K/V GEMMs; dominant PE GEMM runs on
// v_wmma_f32_16x16x4_f32 with LDS-resident weights and online softmax.
// ---------------------------------------------------------------------------

#define CS   256      // channel dim
#define HH   8        // heads
#define DD   32       // head dim
#define KA   204      // 192 (3*64 PE) + 9 (covar) + 3 pad  (multiple of 4)
#define TN   32       // points per tile
#define Y2S  260      // padded Y2 row stride (bank-conflict avoidance)
#define NTH  256      // threads per block (8 waves)

typedef float v2f __attribute__((ext_vector_type(2)));
typedef float v8f __attribute__((ext_vector_type(8)));

__device__ __forceinline__ v8f wmma_f32(v2f a, v2f b, v8f c) {
  // D = A(16x4,f32) * B(4x16,f32) + C(16x16,f32)
  return __builtin_amdgcn_wmma_f32_16x16x4_f32(false, a, false, b,
                                               (short)0, c, false, false);
}

// shared memory layout (floats)
#define OFF_WC   0
#define OFF_AT   (OFF_WC  + KA*CS)     // 52224
#define OFF_Y2   (OFF_AT  + TN*KA)     // +6528
#define OFF_U    (OFF_Y2  + TN*Y2S)    // +8320
#define OFF_Z    (OFF_U   + CS*HH)     // +2048
#define OFF_QV   (OFF_Z   + HH*CS)     // +2048
#define OFF_SL   (OFF_QV  + CS)
#define OFF_LG   (OFF_SL  + CS)
#define OFF_ATT  (OFF_LG  + TN*HH)
#define OFF_RP   (OFF_ATT + TN*HH)
#define OFF_FRQ  (OFF_RP  + TN*3)
#define OFF_RM   (OFF_FRQ + 32)
#define OFF_TV   (OFF_RM  + 12)
#define OFF_MH   (OFF_TV  + 4)
#define OFF_SH   (OFF_MH  + 8)
#define OFF_SCH  (OFF_SH  + 8)
#define OFF_LB   (OFF_SCH + 8)
#define OFF_OUTP (OFF_LB  + 8)
#define SMEM_FLOATS (OFF_OUTP + CS)
#define SMEM_BYTES  (SMEM_FLOATS * 4)  // ~290.5 KB < 320 KB WGP LDS

__global__ __launch_bounds__(NTH)
void ica_fused(const float* __restrict__ s,      const float* __restrict__ r_rot,
               const float* __restrict__ r_trans,const float* __restrict__ x,
               const float* __restrict__ y,      const float* __restrict__ cmat,
               const float* __restrict__ mask,
               const float* __restrict__ W_pos,  const float* __restrict__ b_pos,
               const float* __restrict__ W_covar,const float* __restrict__ b_covar,
               const float* __restrict__ W_kv,   const float* __restrict__ b_kv,
               const float* __restrict__ W_q,    const float* __restrict__ b_q,
               const float* __restrict__ W_out,  const float* __restrict__ b_out,
               const float* __restrict__ gamma,  const float* __restrict__ beta,
               float* __restrict__ out, int N)
{
  extern __shared__ float smem[];
  float* Wc   = smem + OFF_WC;    // [KA][CS]  combined [W_pos; W_covar; 0]
  float* At   = smem + OFF_AT;    // [TN][KA]  A tile: [PE | covar | pad]
  float* Y2   = smem + OFF_Y2;    // [TN][Y2S]
  float* U    = smem + OFF_U;     // [CS][HH]  folded q·W_k
  float* Zs   = smem + OFF_Z;     // [HH][CS]  unnormalized attn·Y2
  float* qv   = smem + OFF_QV;    // [CS]
  float* sl   = smem + OFF_SL;    // [CS]
  float* lgt  = smem + OFF_LG;    // [TN][HH]
  float* att  = smem + OFF_ATT;   // [TN][HH]
  float* rp   = smem + OFF_RP;    // [TN][3]
  float* frq  = smem + OFF_FRQ;   // [32]
  float* Rm   = smem + OFF_RM;    // [9] rotation (row-major R[j][i] at j*3+i)
  float* tv   = smem + OFF_TV;    // [3]
  float* mh   = smem + OFF_MH;    // [8] running max
  float* sh   = smem + OFF_SH;    // [8] running sum
  float* sch  = smem + OFF_SCH;   // [8] rescale factor
  float* lb   = smem + OFF_LB;    // [8] q·b_k per head
  float* outp = smem + OFF_OUTP;  // [CS]

  const int t    = threadIdx.x;
  const int l    = blockIdx.x;
  const int lane = t & 31;
  const int wv   = t >> 5;         // wave id == head id for Z ops
  const int hi   = lane >> 4;      // WMMA half-wave select
  const int lo   = lane & 15;

  // ---- stage combined weight matrix [KA x CS] into LDS ----
  for (int idx = t; idx < KA*CS; idx += NTH) {
    float w;
    if (idx < 192*CS)      w = W_pos[idx];
    else if (idx < 201*CS) w = W_covar[idx - 192*CS];
    else                   w = 0.0f;
    Wc[idx] = w;
  }
  // freqs[j] = exp(-(log(0.1) + j*(log(1000)-log(0.1))/31))
  if (t < 32) frq[t] = expf(2.30258509f - (float)t * 0.29710775f);
  if (t < 9)  Rm[t] = r_rot[l*9 + t];
  if (t < 3)  tv[t] = r_trans[l*3 + t];
  if (t < 8)  { mh[t] = -INFINITY; sh[t] = 0.0f; }
  sl[t] = s[(size_t)l*CS + t];
  __syncthreads();

  // ---- q = s_l @ W_q + b_q ----
  {
    float acc = b_q[t];
    for (int c = 0; c < CS; ++c) acc += sl[c] * W_q[c*CS + t];
    qv[t] = acc;
  }
  __syncthreads();

  // ---- U[cc][h] = sum_dd W_k[cc][h*32+dd] * q[h*32+dd];  lb[h] = q_h·b_k,h ----
  {
    const float* wrow = W_kv + (size_t)t * (2*CS);
    for (int h = 0; h < HH; ++h) {
      float acc = 0.f;
      for (int dd = 0; dd < DD; ++dd) acc += wrow[h*DD + dd] * qv[h*DD + dd];
      U[t*HH + h] = acc;
    }
  }
  if (t < HH) {
    float acc = 0.f;
    for (int dd = 0; dd < DD; ++dd) acc += b_kv[t*DD + dd] * qv[t*DD + dd];
    lb[t] = acc;
  }
  { // zero Z accumulator
    for (int i = 0; i < 8; ++i) Zs[wv*CS + lane + 32*i] = 0.f;
  }
  __syncthreads();

  const float inv_sqrt_d = 0.17677669529663688f;  // 1/sqrt(32)
  const int ntiles = N / TN;

  for (int tile = 0; tile < ntiles; ++tile) {
    const int n0 = tile * TN;

    // ---- relpos + covar for the 32 rows of this tile ----
    if (t < TN) {
      const int n = n0 + t;
      float dx = x[n*3+0]-tv[0], dy = x[n*3+1]-tv[1], dz = x[n*3+2]-tv[2];
      rp[t*3+0] = Rm[0]*dx + Rm[3]*dy + Rm[6]*dz;   // relpos_i = sum_j R[j][i] d_j
      rp[t*3+1] = Rm[1]*dx + Rm[4]*dy + Rm[7]*dz;
      rp[t*3+2] = Rm[2]*dx + Rm[5]*dy + Rm[8]*dz;
      float cm[9], tmp[9];
      for (int j = 0; j < 9; ++j) cm[j] = cmat[(size_t)n*9 + j];
      for (int i = 0; i < 3; ++i)
        for (int k = 0; k < 3; ++k)   // tmp[i][k] = sum_j R[j][i] c[j][k]
          tmp[i*3+k] = Rm[0+i]*cm[0+k] + Rm[3+i]*cm[3+k] + Rm[6+i]*cm[6+k];
      for (int i = 0; i < 3; ++i)
        for (int m = 0; m < 3; ++m)   // covar[i][m] = sum_k tmp[i][k] R[k][m]
          At[t*KA + 192 + i*3 + m] =
              tmp[i*3+0]*Rm[0+m] + tmp[i*3+1]*Rm[3+m] + tmp[i*3+2]*Rm[6+m];
      At[t*KA+201] = 0.f; At[t*KA+202] = 0.f; At[t*KA+203] = 0.f;
    }
    __syncthreads();

    // ---- positional encoding fill: 32 x 192 ----
    for (int e = t; e < TN*192; e += NTH) {
      int row = e / 192, col = e - row*192;
      int i = col >> 6, ch = col & 63;
      float v = rp[row*3 + i] * frq[ch & 31];
      At[row*KA + col] = (ch < 32) ? sinf(v) : cosf(v);
    }
    __syncthreads();

    // ---- WMMA GEMM: Y2[32x256] = At[32xKA] @ Wc[KAx256]  (+ y + biases) ----
    {
      v8f acc00 = {}, acc01 = {}, acc10 = {}, acc11 = {};
      const int colB0 = wv*32 + lo;        // this wave's 32-col slab
      const int colB1 = colB0 + 16;
      const float* a0p = At + lo*KA + 2*hi;        // A rows 0..15
      const float* a1p = At + (16+lo)*KA + 2*hi;   // A rows 16..31
      const float* bp  = Wc + 2*hi*CS;
      for (int k = 0; k < KA; k += 4) {
        v2f a0 = *(const v2f*)(a0p + k);           // A[m][k+2hi .. +1]
        v2f a1 = *(const v2f*)(a1p + k);
        const float* b = bp + k*CS;
        v2f b0 = { b[colB0], b[CS + colB0] };      // B[k+2hi..+1][n]
        v2f b1 = { b[colB1], b[CS + colB1] };
        acc00 = wmma_f32(a0, b0, acc00);
        acc01 = wmma_f32(a0, b1, acc01);
        acc10 = wmma_f32(a1, b0, acc10);
        acc11 = wmma_f32(a1, b1, acc11);
      }
      const float bias0 = b_pos[colB0] + b_covar[colB0];
      const float bias1 = b_pos[colB1] + b_covar[colB1];
      for (int r = 0; r < 8; ++r) {       // D: (vgpr r, lane) -> M=r+8*hi, N=lo
        int row0 = r + 8*hi, row1 = 16 + row0;
        Y2[row0*Y2S + colB0] = acc00[r] + bias0 + y[(size_t)(n0+row0)*CS + colB0];
        Y2[row0*Y2S + colB1] = acc01[r] + bias1 + y[(size_t)(n0+row0)*CS + colB1];
        Y2[row1*Y2S + colB0] = acc10[r] + bias0 + y[(size_t)(n0+row1)*CS + colB0];
        Y2[row1*Y2S + colB1] = acc11[r] + bias1 + y[(size_t)(n0+row1)*CS + colB1];
      }
    }
    __syncthreads();

    // ---- logits: row r = t>>3, head h = t&7 ----
    float lg;
    {
      const int r = t >> 3, h = t & 7;
      float acc = 0.f;
      const float* yr = Y2 + r*Y2S;
      for (int cc = 0; cc < CS; ++cc) acc += yr[cc] * U[cc*HH + h];
      lg = (acc + lb[h]) * inv_sqrt_d + (mask[n0 + r] - 1.0f) * 1e10f;
      lgt[r*HH + h] = lg;
    }
    __syncthreads();

    // ---- online softmax: per-head new max + rescale factor ----
    if (t < HH) {
      float m = mh[t], mt = -INFINITY;
      for (int r = 0; r < TN; ++r) mt = fmaxf(mt, lgt[r*HH + t]);
      float mn = fmaxf(m, mt);
      sch[t] = expf(m - mn);   // 0 when m was -inf
      mh[t]  = mn;
    }
    __syncthreads();

    { // p = exp(logit - max), and rescale Z by sch[h]
      const int r = t >> 3, h = t & 7;
      att[r*HH + h] = expf(lg - mh[h]);
    }
    {
      const float sc = sch[wv];
      for (int i = 0; i < 8; ++i) Zs[wv*CS + lane + 32*i] *= sc;
    }
    __syncthreads();

    if (t < HH) { // running denominator
      float acc = 0.f;
      for (int r = 0; r < TN; ++r) acc += att[r*HH + t];
      sh[t] = sh[t]*sch[t] + acc;
    }
    { // Z[h][c] += sum_r p[r][h] * Y2[r][c]   (h = wave id)
      float zacc[8];
      for (int i = 0; i < 8; ++i) zacc[i] = 0.f;
      for (int r = 0; r < TN; ++r) {
        const float p = att[r*HH + wv];
        const float* yr = Y2 + r*Y2S + lane;
        for (int i = 0; i < 8; ++i) zacc[i] += p * yr[32*i];
      }
      for (int i = 0; i < 8; ++i) Zs[wv*CS + lane + 32*i] += zacc[i];
    }
    __syncthreads();
  }

  // ---- out_pre[hc] = b_v[hc] + (Z[h]·W_v[:,hc]) / s_h ----
  {
    const int h = t >> 5;
    const float inv = 1.0f / sh[h];
    float acc = 0.f;
    const float* wvp = W_kv + CS + t;        // W_kv[cc*512 + 256 + t]
    const float* zr  = Zs + h*CS;
    for (int cc = 0; cc < CS; ++cc) acc += zr[cc] * wvp[(size_t)cc * (2*CS)];
    outp[t] = b_kv[CS + t] + acc * inv;
  }
  __syncthreads();

  // ---- final projection + beta*tanh(gamma) ----
  {
    float acc = b_out[t];
    for (int hc = 0; hc < CS; ++hc) acc += outp[hc] * W_out[hc*CS + t];
    const float scale = beta[0] * tanhf(gamma[0]);
    out[(size_t)l*CS + t] = scale * acc;
  }
}

extern "C" void kernel_launch(void* const* d_in, const int* in_sizes, int n_in,
                              void* d_out, int out_size, void* d_ws, size_t ws_size,
                              hipStream_t stream) {
  const float* s       = (const float*)d_in[0];
  const float* r_rot   = (const float*)d_in[1];
  const float* r_trans = (const float*)d_in[2];
  const float* x       = (const float*)d_in[3];
  const float* y       = (const float*)d_in[4];
  const float* cmat    = (const float*)d_in[5];
  const float* mask    = (const float*)d_in[6];
  const float* W_pos   = (const float*)d_in[7];
  const float* b_pos   = (const float*)d_in[8];
  const float* W_covar = (const float*)d_in[9];
  const float* b_covar = (const float*)d_in[10];
  const float* W_kv    = (const float*)d_in[11];
  const float* b_kv    = (const float*)d_in[12];
  const float* W_q     = (const float*)d_in[13];
  const float* b_q     = (const float*)d_in[14];
  const float* W_out   = (const float*)d_in[15];
  const float* b_out   = (const float*)d_in[16];
  const float* gamma   = (const float*)d_in[17];
  const float* beta    = (const float*)d_in[18];

  const int L = in_sizes[0] / CS;     // B=1
  const int N = in_sizes[3] / 3;      // x is (B,N,3)

  ica_fused<<<dim3(L), dim3(NTH), SMEM_BYTES, stream>>>(
      s, r_rot, r_trans, x, y, cmat, mask,
      W_pos, b_pos, W_covar, b_covar, W_kv, b_kv, W_q, b_q, W_out, b_out,
      gamma, beta, (float*)d_out, N);
}